// SNN_dense_74234214744622
// MI455X (gfx1250) — compile-verified
//
#include <hip/hip_runtime.h>
#include <hip/hip_bf16.h>

// ---------------------------------------------------------------------------
// TTFS SNN dense layer for MI455X (gfx1250, wave32).
//
// Roofline: W (4.2 MB) lives in L2 (192 MB); ~270 MB of L2 reads, ~0.4 GFLOP
// of scan math -> VALU/latency bound. Prefix-sum structure
// (CW = LowerTri x PermutedW) runs on exact-fp32 V_WMMA_F32_16X16X4_F32
// chains; candidate evaluation is VALU with v_rcp fast division.
// ---------------------------------------------------------------------------

#define IN_SIZE   1024
#define OUT_SIZE  1024
#define BATCH     64
#define KTOT      1025            // IN_SIZE + bias
#define MAXT      100000.0f
#define EPSF      1e-10f
#define SORT_N    2048            // padded power of two for bitonic sort
#define XS_STRIDE 1056            // sorted-row stride (>= 1041, 16-aligned)

typedef __attribute__((ext_vector_type(2))) float v2f;
typedef __attribute__((ext_vector_type(8))) float v8f;

// ---------------------------------------------------------------------------
// Kernel 1: per-batch-row bitonic sort of spike times (+ bias @ t=1.0).
// ---------------------------------------------------------------------------
__global__ __launch_bounds__(256)
void snn_sort_kernel(const float* __restrict__ X,
                     float* __restrict__ xs_out,
                     int*   __restrict__ idx_out) {
    __shared__ float key[SORT_N];
    __shared__ int   val[SORT_N];
    const int b = blockIdx.x;
    const int t = threadIdx.x;

    for (int i = t; i < SORT_N; i += 256) {
        float k;
        if (i < IN_SIZE)       k = X[b * IN_SIZE + i];
        else if (i == IN_SIZE) k = 1.0f;        // bias spike time
        else                   k = MAXT;        // pad
        key[i] = k;
        val[i] = i;
    }
    __syncthreads();

    for (int size = 2; size <= SORT_N; size <<= 1) {
        for (int stride = size >> 1; stride > 0; stride >>= 1) {
            for (int i = t; i < SORT_N / 2; i += 256) {
                const int ix = ((i & ~(stride - 1)) << 1) | (i & (stride - 1));
                const int pa = ix;
                const int pb = ix + stride;
                const bool up = ((ix & size) == 0);
                const float ka = key[pa], kb = key[pb];
                const int   va = val[pa], vb = val[pb];
                const bool swap = up ? (ka > kb) : (ka < kb);
                if (swap) {
                    key[pa] = kb; key[pb] = ka;
                    val[pa] = vb; val[pb] = va;
                }
            }
            __syncthreads();
        }
    }

    for (int i = t; i < XS_STRIDE; i += 256) {
        xs_out[b * XS_STRIDE + i]  = key[i];   // positions >= KTOT are MAXT pads
        idx_out[b * XS_STRIDE + i] = val[i];
    }
}

// ---------------------------------------------------------------------------
// One 16-k block of the scan: 8 chained f32 WMMAs + candidate eval.
// GUARD=true only for the final partial block (k0 = 1024): out-of-range
// contributions are removed by zeroing the A-matrix column (exact) and the
// weight-row index is clamped so loads stay in bounds without EXEC divergence.
// ---------------------------------------------------------------------------
template <bool GUARD>
__device__ __forceinline__
void snn_block(int k0,
               const float* __restrict__ Wcol,
               const float* __restrict__ xs,
               const int*   __restrict__ idx,
               int row, int hi,
               const float* amx, const float* amy,   // hoisted lower-tri masks
               float& cwCarry, float& cwtCarry, float& best) {
    v8f dcw, dcwt;
#pragma unroll
    for (int r = 0; r < 8; ++r) { dcw[r] = cwCarry; dcwt[r] = cwtCarry; }

#pragma unroll
    for (int c = 0; c < 4; ++c) {
        const int j0 = 4 * c + 2 * hi;    // K index within block for this lane
        const int kA = k0 + j0;

        float mx = amx[c], my = amy[c];
        int ka0 = kA, ka1 = kA + 1;
        if (GUARD) {
            if (ka0 >= KTOT) { mx = 0.0f; ka0 = KTOT - 1; }
            if (ka1 >= KTOT) { my = 0.0f; ka1 = KTOT - 1; }
        }

        v2f acw, acwt, bfrag;
        acw.x  = mx;
        acw.y  = my;
        acwt.x = mx * xs[kA];             // mask*time: exact where mask==1
        acwt.y = my * xs[kA + 1];
        bfrag.x = Wcol[(size_t)idx[ka0] * OUT_SIZE];
        bfrag.y = Wcol[(size_t)idx[ka1] * OUT_SIZE];

        dcw  = __builtin_amdgcn_wmma_f32_16x16x4_f32(
                   false, acw,  false, bfrag, (short)0, dcw,  false, false);
        dcwt = __builtin_amdgcn_wmma_f32_16x16x4_f32(
                   false, acwt, false, bfrag, (short)0, dcwt, false, false);
    }

    // Candidate evaluation straight out of the C/D register layout.
#pragma unroll
    for (int r = 0; r < 8; ++r) {
        const int   kk   = k0 + r + 8 * hi;      // absolute sorted position
        const float cw   = dcw[r];
        const float cwt  = dcwt[r];
        const float xsk  = xs[kk];
        const float xsk1 = xs[kk + 1];
        const float denom = fminf(fmaxf(cw - 1.0f, EPSF), MAXT);  // v_med3
        const float cand  = cwt * __builtin_amdgcn_rcpf(denom);   // fast div
        const bool ok = (cw >= 1.0f) && (cand >= xsk) && (cand <= xsk1);
        best = fminf(best, ok ? cand : MAXT);
    }

    // Carry = full block sum = prefix row 15, held by hi-lanes in VGPR 7.
    const float tcw  = dcw[7];
    const float tcwt = dcwt[7];
    const float ocw  = __shfl_xor(tcw,  16, 32);
    const float ocwt = __shfl_xor(tcwt, 16, 32);
    cwCarry  = hi ? tcw  : ocw;
    cwtCarry = hi ? tcwt : ocwt;
}

// ---------------------------------------------------------------------------
// Kernel 2: WMMA prefix-sum scan.
// grid = (4 column groups, 64 batch rows), block = 512 threads = 16 waves.
// Each wave owns a 16-column tile; walks 65 k-blocks of 16 sorted inputs.
// ---------------------------------------------------------------------------
__global__ __launch_bounds__(512)
void snn_scan_kernel(const float* __restrict__ W,
                     const float* __restrict__ xs_ws,
                     const int*   __restrict__ idx_ws,
                     float* __restrict__ out) {
    __shared__ float xs[XS_STRIDE];
    __shared__ int   idx[XS_STRIDE];

    const int b      = blockIdx.y;
    const int colgrp = blockIdx.x;          // 0..3
    const int t      = threadIdx.x;

    for (int i = t; i < XS_STRIDE; i += 512) {
        xs[i]  = xs_ws[b * XS_STRIDE + i];
        idx[i] = idx_ws[b * XS_STRIDE + i];
    }
    __syncthreads();

    const int wave = t >> 5;                // 0..15
    const int lane = t & 31;
    const int row  = lane & 15;             // A-matrix M / D-matrix N
    const int hi   = lane >> 4;             // which K-half this lane holds
    const int m0   = colgrp * 256 + wave * 16;
    const float* __restrict__ Wcol = W + m0 + row;   // this lane's column

    // Loop-invariant lower-triangular A masks (per chunk, per lane).
    float amx[4], amy[4];
#pragma unroll
    for (int c = 0; c < 4; ++c) {
        const int j0 = 4 * c + 2 * hi;
        amx[c] = (j0     <= row) ? 1.0f : 0.0f;
        amy[c] = (j0 + 1 <= row) ? 1.0f : 0.0f;
    }

    float cwCarry = 0.0f, cwtCarry = 0.0f;
    float best = MAXT;

    // 64 full (unguarded) blocks, then one guarded partial block.
    for (int k0 = 0; k0 < 1024; k0 += 16)
        snn_block<false>(k0, Wcol, xs, idx, row, hi, amx, amy,
                         cwCarry, cwtCarry, best);
    snn_block<true>(1024, Wcol, xs, idx, row, hi, amx, amy,
                    cwCarry, cwtCarry, best);

    // Min over the two row-halves of the tile sharing the same output column.
    best = fminf(best, __shfl_xor(best, 16, 32));
    if (hi == 0) out[b * OUT_SIZE + m0 + row] = best;
}

// ---------------------------------------------------------------------------
extern "C" void kernel_launch(void* const* d_in, const int* in_sizes, int n_in,
                              void* d_out, int out_size, void* d_ws, size_t ws_size,
                              hipStream_t stream) {
    const float* X = (const float*)d_in[0];   // [64, 1024]
    const float* W = (const float*)d_in[1];   // [1025, 1024]
    float* out = (float*)d_out;               // [64, 1024]

    float* xs_ws  = (float*)d_ws;                           // 64*1056 floats
    int*   idx_ws = (int*)((char*)d_ws +
                           (size_t)BATCH * XS_STRIDE * sizeof(float));

    snn_sort_kernel<<<dim3(BATCH), dim3(256), 0, stream>>>(X, xs_ws, idx_ws);

    snn_scan_kernel<<<dim3(4, BATCH), dim3(512), 0, stream>>>(W, xs_ws, idx_ws, out);
}